// Reservior_predict_8005819040423
// MI455X (gfx1250) — compile-verified
//
#include <hip/hip_runtime.h>

typedef __attribute__((ext_vector_type(2))) float v2f;
typedef __attribute__((ext_vector_type(4))) float v4f;
typedef __attribute__((ext_vector_type(8))) float v8f;

#define N_ROWS 16384
#define M_ROWS 8192
#define D 512

// ---------------------------------------------------------------------------
// K1: v[0..511] = W1 @ W3, v[512..1023] = W2 @ W3   (one wave per output row)
// ---------------------------------------------------------------------------
__global__ void k1_fold_w(const float* __restrict__ W1,
                          const float* __restrict__ W2,
                          const float* __restrict__ W3,
                          float* __restrict__ v) {
  const int wave = (blockIdx.x * blockDim.x + threadIdx.x) >> 5;
  const int lane = threadIdx.x & 31;
  const float* W = (wave < D) ? W1 : W2;
  const int row = wave & (D - 1);
  const float* wr = W + (size_t)row * D;

  float acc = 0.f;
  #pragma unroll 4
  for (int j = lane; j < D; j += 32)
    acc += wr[j] * W3[j];

  #pragma unroll
  for (int off = 16; off > 0; off >>= 1)
    acc += __shfl_xor(acc, off, 32);

  if (lane == 0) v[wave] = acc;
}

// ---------------------------------------------------------------------------
// K2: s1[n] = dot(node[n], v1) ; s2b[m] = dot(sub[m], v2) + b3
// One wave -> 16 rows via V_WMMA_F32_16X16X4_F32.
//   A (16x4 f32): lanes 0-15 hold (M=lane, K=0..1), lanes 16-31 (M=lane-16, K=2..3)
//   B (4x16 f32): v replicated across all 16 columns -> every C column == s[row]
//   C (16x16 f32): VGPR r = row M=r (lanes 0-15) / M=r+8 (lanes 16-31)
// ---------------------------------------------------------------------------
__global__ void k2_gemv_wmma(const float* __restrict__ Xn,   // [16384,512]
                             const float* __restrict__ Xs,   // [ 8192,512]
                             const float* __restrict__ v,    // [1024] = v1|v2
                             const float* __restrict__ b3,   // [1]
                             float* __restrict__ s1,
                             float* __restrict__ s2b) {
  const int wave = (blockIdx.x * blockDim.x + threadIdx.x) >> 5;
  const int lane = threadIdx.x & 31;
  const int NW1  = N_ROWS / 16;                 // 1024 node waves
  const bool isNode = (wave < NW1);

  const float* X  = isNode ? Xn : Xs;
  const float* vv = isNode ? v : (v + D);
  const int row0  = (isNode ? wave : wave - NW1) * 16;

  const int r  = lane & 15;
  const int hi = lane >> 4;                     // 0: K=0,1   1: K=2,3
  const float* xrow = X + (size_t)(row0 + r) * D + hi * 2;

  v8f c = {};
  #pragma unroll 8
  for (int k0 = 0; k0 < D; k0 += 4) {
    v2f a = *(const v2f*)(xrow + k0);           // (K=hi*2, K=hi*2+1) for row r
    v2f b;
    b.x = vv[k0 + hi * 2];                      // B[K=hi*2,   n] = v[k]
    b.y = vv[k0 + hi * 2 + 1];                  // B[K=hi*2+1, n] = v[k]
    c = __builtin_amdgcn_wmma_f32_16x16x4_f32(
        /*neg_a=*/false, a, /*neg_b=*/false, b,
        /*c_mod=*/(short)0, c, /*reuse_a=*/false, /*reuse_b=*/false);
  }

  const float bias = isNode ? 0.f : b3[0];
  float* out = isNode ? s1 : s2b;
  if (r == 0) {                                 // lanes 0 and 16
    const int base = row0 + hi * 8;             // M=r (hi=0) / M=r+8 (hi=1)
    #pragma unroll
    for (int i = 0; i < 8; ++i)
      out[base + i] = c[i] + bias;
  }
}

// ---------------------------------------------------------------------------
// K3: out[n,m] = s1[n] + s2b[m]  — the 536 MB streaming-store kernel.
// Block = 256 threads covers 1024 columns (float4/thread) x 8 rows.
// s2b chunk (4 KB) is loaded once per block and reused across 8 rows;
// stores are non-temporal b128 (output >> L2, write-through streaming).
// ---------------------------------------------------------------------------
__global__ void k3_outer_sum(const float* __restrict__ s1,
                             const float* __restrict__ s2b,
                             float* __restrict__ out) {
  const int c4   = blockIdx.x * blockDim.x + threadIdx.x;   // 0..2047 (col/4)
  const int row0 = blockIdx.y * 8;

  const v4f t = ((const v4f*)s2b)[c4];
  const v4f sA = *(const v4f*)(s1 + row0);       // rows row0..row0+3 (uniform)
  const v4f sB = *(const v4f*)(s1 + row0 + 4);   // rows row0+4..row0+7

  v4f* o = (v4f*)out + (size_t)row0 * (M_ROWS / 4) + c4;
  #pragma unroll
  for (int i = 0; i < 4; ++i) {
    __builtin_nontemporal_store(t + sA[i], o);
    o += (M_ROWS / 4);
  }
  #pragma unroll
  for (int i = 0; i < 4; ++i) {
    __builtin_nontemporal_store(t + sB[i], o);
    o += (M_ROWS / 4);
  }
}

// ---------------------------------------------------------------------------
extern "C" void kernel_launch(void* const* d_in, const int* in_sizes, int n_in,
                              void* d_out, int out_size, void* d_ws, size_t ws_size,
                              hipStream_t stream) {
  const float* node = (const float*)d_in[0];   // [16384,512]
  const float* sub  = (const float*)d_in[1];   // [ 8192,512]
  const float* W1   = (const float*)d_in[2];   // [512,512]
  const float* W2   = (const float*)d_in[3];   // [512,512]
  const float* W3   = (const float*)d_in[4];   // [512,1]
  const float* b3   = (const float*)d_in[5];   // [1]
  float* out        = (float*)d_out;           // [16384,8192]

  float* v   = (float*)d_ws;                   // 1024 floats  (v1 | v2)
  float* s1  = v + 2 * D;                      // 16384 floats
  float* s2b = s1 + N_ROWS;                    // 8192 floats

  // K1: 1024 waves (one per folded-weight row) -> 128 blocks x 256
  k1_fold_w<<<dim3(128), dim3(256), 0, stream>>>(W1, W2, W3, v);

  // K2: 1024 + 512 = 1536 waves -> 192 blocks x 256
  k2_gemv_wmma<<<dim3(192), dim3(256), 0, stream>>>(node, sub, v, b3, s1, s2b);

  // K3: grid (8192/1024 col-chunks, 16384/8 row-groups) x 256 threads
  k3_outer_sum<<<dim3(M_ROWS / 1024, N_ROWS / 8), dim3(256), 0, stream>>>(s1, s2b, out);
}